// DNNPerf_88510686036316
// MI455X (gfx1250) — compile-verified
//
#include <hip/hip_runtime.h>
#include <hip/hip_bf16.h>
#include <math.h>

#define NNODES 50000
#define NEDGES 600000
#define DH 128
#define DE 64
#define ALPHA 0.01f

typedef float v2f __attribute__((ext_vector_type(2)));
typedef float v8f __attribute__((ext_vector_type(8)));

__device__ __forceinline__ float lrelu(float x) { return x > 0.f ? x : ALPHA * x; }

// ---------------------------------------------------------------------------
// Kernel 0: w_em[i] = sum_k W_e[i,k] * W_m[k]   (collapses e@W_e@W_m chain)
// ---------------------------------------------------------------------------
__global__ void precompute_wem(const float* __restrict__ We,
                               const float* __restrict__ Wm,
                               float* __restrict__ wem) {
  int i = threadIdx.x;
  if (i < DE) {
    float s = 0.f;
    for (int k = 0; k < DH; ++k) s += We[i * DH + k] * Wm[k];
    wem[i] = s;
  }
}

// ---------------------------------------------------------------------------
// Kernel 1: h_prime = leakyrelu(h @ W_u)  via V_WMMA_F32_16X16X4_F32
// One wave per 16-row strip; 8 column tiles of 16; W_u staged in LDS (64KB).
// ---------------------------------------------------------------------------
__global__ __launch_bounds__(256) void node_gemm_wmma(
    const float* __restrict__ h, const float* __restrict__ Wu,
    float* __restrict__ hp) {
  __shared__ float Wlds[DH * DH];  // 64 KB
  const int tid = threadIdx.x;

  // cooperative coalesced load of W_u into LDS (float4)
  {
    const float4* s4 = (const float4*)Wu;
    float4* d4 = (float4*)Wlds;
    for (int i = tid; i < DH * DH / 4; i += 256) d4[i] = s4[i];
  }
  __syncthreads();

  const int wave = tid >> 5;
  const int lane = tid & 31;
  const int strip = blockIdx.x * 8 + wave;
  if (strip >= NNODES / 16) return;  // whole-wave uniform: EXEC stays all-1s

  const int row0 = strip * 16;
  const int m = lane & 15;    // M (lanes 0-15) / M (lanes 16-31)
  const int kh = lane >> 4;   // K-half select: {0,1} vs {2,3}

  v8f acc[8] = {};
  const float* arow = h + (size_t)(row0 + m) * DH;

  for (int kk = 0; kk < DH; kk += 4) {
    v2f a;
    a.x = arow[kk + 2 * kh + 0];   // A[m][kk + 2*kh + 0]
    a.y = arow[kk + 2 * kh + 1];   // A[m][kk + 2*kh + 1]
    const float* wr0 = &Wlds[(kk + 2 * kh + 0) * DH];
    const float* wr1 = &Wlds[(kk + 2 * kh + 1) * DH];
#pragma unroll
    for (int nt = 0; nt < 8; ++nt) {
      v2f b;
      b.x = wr0[nt * 16 + m];      // B[kk + 2*kh + 0][n]
      b.y = wr1[nt * 16 + m];      // B[kk + 2*kh + 1][n]
      acc[nt] = __builtin_amdgcn_wmma_f32_16x16x4_f32(
          false, a, false, b, (short)0, acc[nt], false, false);
    }
  }

  // C/D layout: VGPR i -> row (row0 + i + 8*kh), col = nt*16 + m
#pragma unroll
  for (int nt = 0; nt < 8; ++nt) {
#pragma unroll
    for (int i = 0; i < 8; ++i) {
      int row = row0 + i + 8 * kh;
      int col = nt * 16 + m;
      hp[(size_t)row * DH + col] = lrelu(acc[nt][i]);
    }
  }
}

// ---------------------------------------------------------------------------
// Kernel 2: per-edge logit  z_e = sigmoid(hs·a_lo + hd·a_hi) * (e_row · w_em)
// One wave32 per edge: float4 gathers of h_prime rows (L2-resident).
// ---------------------------------------------------------------------------
__global__ __launch_bounds__(256) void edge_score(
    const float* __restrict__ hp, const float* __restrict__ e,
    const int* __restrict__ src, const int* __restrict__ dst,
    const float* __restrict__ a, const float* __restrict__ wem,
    float* __restrict__ z, int nWaves) {
  const int lane = threadIdx.x & 31;
  const int wid = (blockIdx.x * blockDim.x + threadIdx.x) >> 5;

  const float4 al = ((const float4*)a)[lane];          // a[0:128]
  const float4 ah = ((const float4*)(a + DH))[lane];   // a[128:256]
  const float2 wm2 = ((const float2*)wem)[lane];

  for (int eid = wid; eid < NEDGES; eid += nWaves) {
    const int s = src[eid];
    const int d = dst[eid];
    const float4 hs = ((const float4*)(hp + (size_t)s * DH))[lane];
    const float4 hd = ((const float4*)(hp + (size_t)d * DH))[lane];
    float ps = hs.x * al.x + hs.y * al.y + hs.z * al.z + hs.w * al.w +
               hd.x * ah.x + hd.y * ah.y + hd.z * ah.z + hd.w * ah.w;
    const float2 ev = ((const float2*)(e + (size_t)eid * DE))[lane];
    float pw = ev.x * wm2.x + ev.y * wm2.y;
#pragma unroll
    for (int off = 16; off > 0; off >>= 1) {
      ps += __shfl_xor(ps, off, 32);
      pw += __shfl_xor(pw, off, 32);
    }
    if (lane == 0) {
      float sig = 1.f / (1.f + __expf(-ps));
      z[eid] = sig * pw;
    }
  }
}

// ---------------------------------------------------------------------------
// Softmax statistics (two-pass over the 2.4 MB z array, L2-resident)
// ---------------------------------------------------------------------------
__global__ __launch_bounds__(256) void reduce_max(const float* __restrict__ z,
                                                  float* __restrict__ pmax) {
  __shared__ float sm[256];
  float m = -1e30f;
  for (int i = blockIdx.x * 256 + threadIdx.x; i < NEDGES; i += gridDim.x * 256)
    m = fmaxf(m, z[i]);
  sm[threadIdx.x] = m;
  __syncthreads();
  for (int s = 128; s > 0; s >>= 1) {
    if (threadIdx.x < s) sm[threadIdx.x] = fmaxf(sm[threadIdx.x], sm[threadIdx.x + s]);
    __syncthreads();
  }
  if (threadIdx.x == 0) pmax[blockIdx.x] = sm[0];
}

__global__ __launch_bounds__(256) void reduce_final_max(const float* __restrict__ pmax,
                                                        float* __restrict__ stats) {
  __shared__ float sm[256];
  sm[threadIdx.x] = pmax[threadIdx.x];
  __syncthreads();
  for (int s = 128; s > 0; s >>= 1) {
    if (threadIdx.x < s) sm[threadIdx.x] = fmaxf(sm[threadIdx.x], sm[threadIdx.x + s]);
    __syncthreads();
  }
  if (threadIdx.x == 0) stats[0] = sm[0];
}

__global__ __launch_bounds__(256) void reduce_sumexp(const float* __restrict__ z,
                                                     const float* __restrict__ stats,
                                                     float* __restrict__ psum) {
  __shared__ float sm[256];
  const float gmax = stats[0];
  float s = 0.f;
  for (int i = blockIdx.x * 256 + threadIdx.x; i < NEDGES; i += gridDim.x * 256)
    s += __expf(z[i] - gmax);
  sm[threadIdx.x] = s;
  __syncthreads();
  for (int t = 128; t > 0; t >>= 1) {
    if (threadIdx.x < t) sm[threadIdx.x] += sm[threadIdx.x + t];
    __syncthreads();
  }
  if (threadIdx.x == 0) psum[blockIdx.x] = sm[0];
}

__global__ __launch_bounds__(256) void reduce_final_sum(const float* __restrict__ psum,
                                                        float* __restrict__ stats,
                                                        float* __restrict__ hg) {
  __shared__ float sm[256];
  sm[threadIdx.x] = psum[threadIdx.x];
  __syncthreads();
  for (int t = 128; t > 0; t >>= 1) {
    if (threadIdx.x < t) sm[threadIdx.x] += sm[threadIdx.x + t];
    __syncthreads();
  }
  if (threadIdx.x == 0) stats[1] = 1.f / sm[0];
  if (threadIdx.x < DH) hg[threadIdx.x] = 0.f;  // zero accumulator each launch
}

// ---------------------------------------------------------------------------
// Kernel 4: hg[c] = sum_e leakyrelu(sm_e * h_prime[src[e], c])
// (segment_sum + node-sum fused; scatter eliminated)
// ---------------------------------------------------------------------------
__global__ __launch_bounds__(128) void edge_accum(
    const float* __restrict__ hp, const int* __restrict__ src,
    const float* __restrict__ z, const float* __restrict__ stats,
    float* __restrict__ hg) {
  const float gmax = stats[0];
  const float invZ = stats[1];
  const int t = threadIdx.x;
  float acc = 0.f;
  for (int eid = blockIdx.x; eid < NEDGES; eid += gridDim.x) {
    const float smv = __expf(z[eid] - gmax) * invZ;  // >= 0
    const float v = smv * hp[(size_t)src[eid] * DH + t];
    acc += (v > 0.f) ? v : ALPHA * v;
  }
  atomicAdd(&hg[t], acc);
}

// ---------------------------------------------------------------------------
// Kernel 5: readout MLP 128 -> 512 -> 128 -> 16 -> 1 (single block)
// ---------------------------------------------------------------------------
__global__ __launch_bounds__(512) void readout_mlp(
    const float* __restrict__ hg,
    const float* __restrict__ W1, const float* __restrict__ b1,
    const float* __restrict__ W2, const float* __restrict__ b2,
    const float* __restrict__ W3, const float* __restrict__ b3,
    const float* __restrict__ W4, const float* __restrict__ b4,
    float* __restrict__ out) {
  __shared__ float x0[DH], x1[512], x2[DH], x3[16];
  const int t = threadIdx.x;
  if (t < DH) x0[t] = hg[t];
  __syncthreads();
  {
    float s = b1[t];
    for (int k = 0; k < DH; ++k) s += x0[k] * W1[k * 512 + t];
    x1[t] = fmaxf(s, 0.f);
  }
  __syncthreads();
  if (t < DH) {
    float s = b2[t];
    for (int k = 0; k < 512; ++k) s += x1[k] * W2[k * DH + t];
    x2[t] = fmaxf(s, 0.f);
  }
  __syncthreads();
  if (t < 16) {
    float s = b3[t];
    for (int k = 0; k < DH; ++k) s += x2[k] * W3[k * 16 + t];
    x3[t] = fmaxf(s, 0.f);
  }
  __syncthreads();
  if (t == 0) {
    float s = b4[0];
    for (int k = 0; k < 16; ++k) s += x3[k] * W4[k];
    out[0] = s;
  }
}

// ---------------------------------------------------------------------------
extern "C" void kernel_launch(void* const* d_in, const int* in_sizes, int n_in,
                              void* d_out, int out_size, void* d_ws, size_t ws_size,
                              hipStream_t stream) {
  (void)in_sizes; (void)n_in; (void)out_size; (void)ws_size;
  const float* h  = (const float*)d_in[0];
  const float* e  = (const float*)d_in[1];
  const int*   src = (const int*)d_in[2];
  const int*   dst = (const int*)d_in[3];
  const float* Wu = (const float*)d_in[4];
  const float* We = (const float*)d_in[5];
  const float* a  = (const float*)d_in[6];
  const float* Wm = (const float*)d_in[7];
  const float* W1 = (const float*)d_in[8];
  const float* b1 = (const float*)d_in[9];
  const float* W2 = (const float*)d_in[10];
  const float* b2 = (const float*)d_in[11];
  const float* W3 = (const float*)d_in[12];
  const float* b3 = (const float*)d_in[13];
  const float* W4 = (const float*)d_in[14];
  const float* b4 = (const float*)d_in[15];

  float* ws = (float*)d_ws;
  float* hp    = ws;                 // 6,400,000 f  (h_prime, L2-resident 25.6 MB)
  float* z     = ws + 6400000;       //   600,000 f  (edge logits)
  float* wem   = ws + 7000000;       //        64 f
  float* pmax  = ws + 7000064;       //       256 f
  float* psum  = ws + 7000320;       //       256 f
  float* stats = ws + 7000576;       //         2 f  (gmax, 1/Z)
  float* hg    = ws + 7000578;       //       128 f

  precompute_wem<<<1, 64, 0, stream>>>(We, Wm, wem);
  node_gemm_wmma<<<(NNODES / 16 + 7) / 8, 256, 0, stream>>>(h, Wu, hp);
  edge_score<<<1024, 256, 0, stream>>>(hp, e, src, dst, a, wem, z, 1024 * 256 / 32);
  reduce_max<<<256, 256, 0, stream>>>(z, pmax);
  reduce_final_max<<<1, 256, 0, stream>>>(pmax, stats);
  reduce_sumexp<<<256, 256, 0, stream>>>(z, stats, psum);
  reduce_final_sum<<<1, 256, 0, stream>>>(psum, stats, hg);
  edge_accum<<<1024, 128, 0, stream>>>(hp, src, z, stats, hg);
  readout_mlp<<<1, 512, 0, stream>>>(hg, W1, b1, W2, b2, W3, b3, W4, b4,
                                     (float*)d_out);
  (void)dst;
}